// GravNetBlock_53979148976760
// MI455X (gfx1250) — compile-verified
//
#include <hip/hip_runtime.h>
#include <hip/hip_bf16.h>
#include <math.h>

// GravNet stack for MI455X (gfx1250, wave32).
// - All dense / lin_out GEMMs use V_WMMA_F32_16X16X4_F32 (full fp32 precision).
//   Shapes are compile-time template params -> fully unrolled K-loop, constant
//   addressing, one wave per 16-row x FOUT strip (A fragments reused across
//   FOUT/16 accumulators).
// - kNN (k=3) is a streaming top-3 over LDS-staged (s,sq,h) tiles; N x N
//   distances are never materialized (would be 1 GB).
// - agg@w2 + b2 (+ReLU) fused into the WMMA GEMM epilogue.

#define NNODES 16384

typedef float v2f __attribute__((ext_vector_type(2)));
typedef float v8f __attribute__((ext_vector_type(8)));

// ---------------------------------------------------------------------------
// GEMM: out[M,FOUT] = A[M,FIN] @ W[FIN,FOUT] + bias[n]
//       (+ agg[m,0]*w2[0,n] + agg[m,1]*w2[1,n] if AGG)   (+ ReLU if RELU)
// One wave per 16-row strip; FOUT/16 accumulators; K stepped by 4.
// Fragment layouts per CDNA5 ISA 7.12.2:
//   A 16x4 : lane = M (mod 16), VGPR0/1 = K, lane-half selects K+0..1 / K+2..3
//   B 4x16 : lane = N (mod 16), VGPR0/1 = K, lane-half selects K rows 0..1/2..3
//   C 16x16: lane = N (mod 16), VGPR v -> M = v + 8*(lane>=16)
// ---------------------------------------------------------------------------
template <int FIN, int FOUT, bool AGG, bool RELU>
__global__ __launch_bounds__(256) void gemm_wmma_kernel(
    const float* __restrict__ A, const float* __restrict__ W,
    const float* __restrict__ bias, const float* __restrict__ agg,
    const float* __restrict__ w2, float* __restrict__ out)
{
  constexpr int NT = FOUT / 16;
  const int wave = threadIdx.x >> 5;
  const int lane = threadIdx.x & 31;
  const int mt   = blockIdx.x * 8 + wave;   // 16-row strip index
  const int m0   = mt << 4;
  const int half = lane >> 4;               // 0: lanes 0-15, 1: lanes 16-31
  const int lm   = lane & 15;

  // A row pointer pre-offset by this half's K offset (k + 2*half)
  const float* Arow = A + (size_t)(m0 + lm) * FIN + (half << 1);

  v8f c[NT];
#pragma unroll
  for (int t = 0; t < NT; ++t) {
    v8f z = {0.f, 0.f, 0.f, 0.f, 0.f, 0.f, 0.f, 0.f};
    c[t] = z;
  }

#pragma unroll
  for (int k = 0; k < FIN; k += 4) {
    const int kk = k + (half << 1);
    const v2f a = *(const v2f*)(Arow + k);  // A[m][kk], A[m][kk+1] (8B aligned)
#pragma unroll
    for (int t = 0; t < NT; ++t) {
      v2f b;
      b.x = W[kk * FOUT + t * 16 + lm];         // W[kk][n]
      b.y = W[(kk + 1) * FOUT + t * 16 + lm];   // W[kk+1][n]
      c[t] = __builtin_amdgcn_wmma_f32_16x16x4_f32(
          /*neg_a=*/false, a, /*neg_b=*/false, b,
          /*c_mod=*/(short)0, c[t], /*reuse_a=*/false, /*reuse_b=*/false);
    }
  }

  float a0[8], a1[8];
  if constexpr (AGG) {
#pragma unroll
    for (int v = 0; v < 8; ++v) {
      const int m = m0 + v + (half << 3);
      a0[v] = agg[2 * m + 0];
      a1[v] = agg[2 * m + 1];
    }
  }

#pragma unroll
  for (int t = 0; t < NT; ++t) {
    const int n = t * 16 + lm;
    const float bv = bias[n];
    float w20 = 0.f, w21 = 0.f;
    if constexpr (AGG) { w20 = w2[n]; w21 = w2[FOUT + n]; }
#pragma unroll
    for (int v = 0; v < 8; ++v) {
      const int m = m0 + v + (half << 3);
      float val = c[t][v] + bv;
      if constexpr (AGG) {
        val = fmaf(a0[v], w20, val);
        val = fmaf(a1[v], w21, val);
      }
      if constexpr (RELU) val = fmaxf(val, 0.f);
      out[(size_t)m * FOUT + n] = val;
    }
  }
}

// ---------------------------------------------------------------------------
// s = x @ ws[fin,3] + bs ; h = x @ wh[fin,1] + bh  -> packed sh[N][4]
// ---------------------------------------------------------------------------
__global__ __launch_bounds__(256) void sh_kernel(
    const float* __restrict__ x, const float* __restrict__ ws,
    const float* __restrict__ bs, const float* __restrict__ wh,
    const float* __restrict__ bh, float* __restrict__ sh, int fin)
{
  const int i = blockIdx.x * 256 + threadIdx.x;
  const float* xr = x + (size_t)i * fin;
  float s0 = bs[0], s1 = bs[1], s2 = bs[2], h = bh[0];
  for (int k = 0; k < fin; ++k) {
    const float xv = xr[k];
    s0 = fmaf(xv, ws[3 * k + 0], s0);
    s1 = fmaf(xv, ws[3 * k + 1], s1);
    s2 = fmaf(xv, ws[3 * k + 2], s2);
    h  = fmaf(xv, wh[k], h);
  }
  ((float4*)sh)[i] = make_float4(s0, s1, s2, h);
}

// ---------------------------------------------------------------------------
// Brute-force kNN (k=3, self included) + GravNet aggregation.
// d2(i,j) = sq_i + sq_j - 2*dot(s_i,s_j)  (reference formula; self pair == 0.0
// exactly since dot==sq via identical fma chains). Streaming top-3 per target.
// agg[i] = (mean_k, max_k) of h[idx_k] * exp(-10 * max(d2_k, 0)).
// ---------------------------------------------------------------------------
__global__ __launch_bounds__(256) void knn_agg_kernel(
    const float* __restrict__ sh, float* __restrict__ agg)
{
  __shared__ float4 t4[256];   // (sx, sy, sz, sq)
  __shared__ float  th[256];   // h
  const int i = blockIdx.x * 256 + threadIdx.x;
  const float4 me = ((const float4*)sh)[i];
  const float sqi = fmaf(me.x, me.x, fmaf(me.y, me.y, me.z * me.z));

  float bd0 = 3.4e38f, bd1 = 3.4e38f, bd2 = 3.4e38f;
  float bh0 = 0.f, bh1 = 0.f, bh2 = 0.f;

  for (int jb = 0; jb < NNODES; jb += 256) {
    const float4 v = ((const float4*)sh)[jb + threadIdx.x];
    const float sq = fmaf(v.x, v.x, fmaf(v.y, v.y, v.z * v.z));
    __syncthreads();
    t4[threadIdx.x] = make_float4(v.x, v.y, v.z, sq);
    th[threadIdx.x] = v.w;
    __syncthreads();
#pragma unroll 4
    for (int jj = 0; jj < 256; ++jj) {
      const float4 b = t4[jj];                                   // ds_load_b128 broadcast
      const float dot = fmaf(me.x, b.x, fmaf(me.y, b.y, me.z * b.z));
      const float d2  = fmaf(-2.f, dot, sqi + b.w);
      if (d2 < bd2) {                                            // rarely taken
        const float hj = th[jj];
        if (d2 < bd0)      { bd2 = bd1; bh2 = bh1; bd1 = bd0; bh1 = bh0; bd0 = d2; bh0 = hj; }
        else if (d2 < bd1) { bd2 = bd1; bh2 = bh1; bd1 = d2;  bh1 = hj; }
        else               { bd2 = d2;  bh2 = hj; }
      }
    }
  }

  const float w0 = expf(-10.f * fmaxf(bd0, 0.f));
  const float w1 = expf(-10.f * fmaxf(bd1, 0.f));
  const float w2 = expf(-10.f * fmaxf(bd2, 0.f));
  const float m0 = bh0 * w0, m1 = bh1 * w1, m2 = bh2 * w2;
  agg[2 * i + 0] = (m0 + m1 + m2) * (1.0f / 3.0f);
  agg[2 * i + 1] = fmaxf(m0, fmaxf(m1, m2));
}

// ---------------------------------------------------------------------------
template <int FIN, int FOUT, bool AGG, bool RELU>
static void launch_gemm(const float* A, const float* W, const float* bias,
                        const float* agg, const float* w2, float* out,
                        hipStream_t stream)
{
  // One wave per 16-row strip: 1024 strips / 8 waves per block = 128 blocks.
  hipLaunchKernelGGL((gemm_wmma_kernel<FIN, FOUT, AGG, RELU>),
                     dim3(NNODES / 16 / 8), dim3(256), 0, stream,
                     A, W, bias, agg, w2, out);
}

extern "C" void kernel_launch(void* const* d_in, const int* in_sizes, int n_in,
                              void* d_out, int out_size, void* d_ws, size_t ws_size,
                              hipStream_t stream)
{
  (void)in_sizes; (void)n_in; (void)out_size; (void)ws_size;
  const float* x = (const float*)d_in[0];
  auto P = [&](int i) { return (const float*)d_in[i]; };

  // Workspace layout (floats): xA[N*64] | xB[N*64] | sh[N*4] | agg[N*2]  (~8.4 MB)
  float* xA  = (float*)d_ws;
  float* xB  = xA + (size_t)NNODES * 64;
  float* sh  = xB + (size_t)NNODES * 64;
  float* agg = sh + (size_t)NNODES * 4;

  // dense1 (no relu), dense2/3 (relu)
  launch_gemm<64, 64, false, false>(x,  P(1), P(2), nullptr, nullptr, xA, stream);
  launch_gemm<64, 64, false, true >(xA, P(3), P(4), nullptr, nullptr, xB, stream);
  launch_gemm<64, 64, false, true >(xB, P(5), P(6), nullptr, nullptr, xA, stream);

  // GravNet chain: 64->64->64->32->32->16->16
  // mp i param base: b = 13 + 7*i  ->  ws, bs, wh, bh, w1, w2, b2
  const int fins[6] = {64, 64, 64, 32, 32, 16};
  float* cur = xA;
  float* nxt = xB;
  for (int i = 0; i < 6; ++i) {
    const int b = 13 + 7 * i;
    hipLaunchKernelGGL(sh_kernel, dim3(NNODES / 256), dim3(256), 0, stream,
                       cur, P(b), P(b + 1), P(b + 2), P(b + 3), sh, fins[i]);
    hipLaunchKernelGGL(knn_agg_kernel, dim3(NNODES / 256), dim3(256), 0, stream,
                       sh, agg);
    // out = cur @ w1 + agg @ w2 + b2   (lin_out1 has no bias)
    switch (i) {
      case 0: launch_gemm<64, 64, true, false>(cur, P(b+4), P(b+6), agg, P(b+5), nxt, stream); break;
      case 1: launch_gemm<64, 64, true, false>(cur, P(b+4), P(b+6), agg, P(b+5), nxt, stream); break;
      case 2: launch_gemm<64, 32, true, false>(cur, P(b+4), P(b+6), agg, P(b+5), nxt, stream); break;
      case 3: launch_gemm<32, 32, true, false>(cur, P(b+4), P(b+6), agg, P(b+5), nxt, stream); break;
      case 4: launch_gemm<32, 16, true, false>(cur, P(b+4), P(b+6), agg, P(b+5), nxt, stream); break;
      case 5: launch_gemm<16, 16, true, false>(cur, P(b+4), P(b+6), agg, P(b+5), nxt, stream); break;
    }
    float* t = cur; cur = nxt; nxt = t;
  }
  // cur == xA, [N,16]

  // dense4/5/6 (relu)
  launch_gemm<16, 64, false, true>(cur, P(7),  P(8),  nullptr, nullptr, nxt, stream);
  launch_gemm<64, 64, false, true>(nxt, P(9),  P(10), nullptr, nullptr, cur, stream);
  launch_gemm<64, 64, false, true>(cur, P(11), P(12), nullptr, nullptr,
                                   (float*)d_out, stream);
}